// Subtractor25Bit_23407571764112
// MI455X (gfx1250) — compile-verified
//
#include <hip/hip_runtime.h>
#include <stdint.h>

// 25-bit ripple-carry subtractor == 25-bit integer subtract per row.
// Memory-bound: ~608 MB HBM traffic -> ~26 us floor @ 23.3 TB/s.
// CDNA5 paths: async global<->LDS b128 DMA (ASYNCcnt) for coalescing the
// 100-byte row pitch, with NT temporal hints (single-use streaming data,
// don't churn the 192 MB L2). WMMA intentionally unused: zero-FLOP workload,
// matrix ops cannot reduce bytes moved.

#define BITS 25
#define ROWS_PER_BLOCK 128
#define THREADS 128
#define MASK25 0x1FFFFFFu

__device__ __forceinline__ uint32_t lds_off(const void* p) {
    // generic ptr to __shared__: low 32 bits are the LDS byte offset
    return (uint32_t)(uintptr_t)p;
}

__device__ __forceinline__ void async_g2l_b128(uint32_t lds, uint64_t gaddr) {
    asm volatile("global_load_async_to_lds_b128 %0, %1, off th:TH_LOAD_NT"
                 :: "v"(lds), "v"(gaddr) : "memory");
}

__device__ __forceinline__ void async_l2g_b128(uint64_t gaddr, uint32_t lds) {
    asm volatile("global_store_async_from_lds_b128 %0, %1, off th:TH_STORE_NT"
                 :: "v"(gaddr), "v"(lds) : "memory");
}

__device__ __forceinline__ void wait_async0() {
    asm volatile("s_wait_asynccnt 0x0" ::: "memory");
}

__global__ __launch_bounds__(THREADS)
void Subtractor25Bit_kernel(const uint32_t* __restrict__ A,
                            const uint32_t* __restrict__ B,
                            uint32_t* __restrict__ out,
                            long long nrows) {
    __shared__ uint32_t sA[ROWS_PER_BLOCK * BITS];   // 12800 B
    __shared__ uint32_t sB[ROWS_PER_BLOCK * BITS];   // 12800 B
    __shared__ uint32_t sO[ROWS_PER_BLOCK * BITS];   // 12800 B
    __shared__ uint32_t sBor[ROWS_PER_BLOCK];        //   512 B

    const int t = threadIdx.x;
    const long long row0   = (long long)blockIdx.x * ROWS_PER_BLOCK;
    const long long eltOff = row0 * BITS;                       // float index of tile start
    const int rowsInTile   = (int)((nrows - row0) < ROWS_PER_BLOCK
                                   ? (nrows - row0) : ROWS_PER_BLOCK);

    if (rowsInTile == ROWS_PER_BLOCK) {
        // ---- fast path: full tile, async-DMA staged, fully coalesced ----
        const int nf4 = ROWS_PER_BLOCK * BITS / 4;              // 800 float4 per operand
        const uint64_t aAddr = (uint64_t)(uintptr_t)A + (uint64_t)eltOff * 4u;
        const uint64_t bAddr = (uint64_t)(uintptr_t)B + (uint64_t)eltOff * 4u;
        const uint32_t sAo = lds_off(sA);
        const uint32_t sBo = lds_off(sB);

        for (int i = t; i < nf4; i += THREADS) {
            async_g2l_b128(sAo + (uint32_t)i * 16u, aAddr + (uint64_t)i * 16u);
            async_g2l_b128(sBo + (uint32_t)i * 16u, bAddr + (uint64_t)i * 16u);
        }
        wait_async0();          // this wave's DMAs landed in LDS
        __syncthreads();        // everyone's DMAs landed

        // ---- one row per thread; LDS stride 25 (odd) => bank-conflict-free ----
        const int base = t * BITS;
        uint32_t ua = 0u, ub = 0u;
#pragma unroll
        for (int i = 0; i < BITS; ++i) {
            // bit pattern of 1.0f is 0x3F800000; >>29 yields exactly 1 (0.0f -> 0)
            ua |= (sA[base + i] >> 29) << i;
            ub |= (sB[base + i] >> 29) << i;
        }
        const uint32_t diff = (ua - ub) & MASK25;               // A + ~B + 1 (mod 2^25)
#pragma unroll
        for (int i = 0; i < BITS; ++i)
            sO[base + i] = ((diff >> i) & 1u) ? 0x3F800000u : 0u;
        sBor[t] = (ua < ub) ? 0x3F800000u : 0u;                 // borrow = 1 - carry_out

        __syncthreads();        // results visible to the DMA engine

        // ---- async store: result tile (contiguous) + borrow tile ----
        const uint64_t oAddr = (uint64_t)(uintptr_t)out + (uint64_t)eltOff * 4u;
        const uint32_t sOo = lds_off(sO);
        for (int i = t; i < nf4; i += THREADS)
            async_l2g_b128(oAddr + (uint64_t)i * 16u, sOo + (uint32_t)i * 16u);

        const uint64_t borAddr = (uint64_t)(uintptr_t)out
                               + ((uint64_t)nrows * BITS + (uint64_t)row0) * 4u;
        const uint32_t sBorO = lds_off(sBor);
        for (int i = t; i < ROWS_PER_BLOCK / 4; i += THREADS)
            async_l2g_b128(borAddr + (uint64_t)i * 16u, sBorO + (uint32_t)i * 16u);
        // S_ENDPGM performs an implicit wait-idle, draining ASYNCcnt.
    } else {
        // ---- tail path (never taken for BATCH=2M): plain per-row loads ----
        const long long r = row0 + t;
        if (r < nrows) {
            const long long rb = r * BITS;
            uint32_t ua = 0u, ub = 0u;
#pragma unroll
            for (int i = 0; i < BITS; ++i) {
                ua |= (A[rb + i] >> 29) << i;
                ub |= (B[rb + i] >> 29) << i;
            }
            const uint32_t diff = (ua - ub) & MASK25;
#pragma unroll
            for (int i = 0; i < BITS; ++i)
                out[rb + i] = ((diff >> i) & 1u) ? 0x3F800000u : 0u;
            out[(long long)nrows * BITS + r] = (ua < ub) ? 0x3F800000u : 0u;
        }
    }
}

extern "C" void kernel_launch(void* const* d_in, const int* in_sizes, int n_in,
                              void* d_out, int out_size, void* d_ws, size_t ws_size,
                              hipStream_t stream) {
    const uint32_t* A = (const uint32_t*)d_in[0];   // float32 {0,1} bits, read as raw u32
    const uint32_t* B = (const uint32_t*)d_in[1];
    uint32_t* out = (uint32_t*)d_out;               // float32 outputs, written as raw u32

    const long long nrows = (long long)in_sizes[0] / BITS;      // 2,000,000
    const int blocks = (int)((nrows + ROWS_PER_BLOCK - 1) / ROWS_PER_BLOCK); // 15625

    Subtractor25Bit_kernel<<<dim3(blocks), dim3(THREADS), 0, stream>>>(A, B, out, nrows);
}